// Affinity_81449759802117
// MI455X (gfx1250) — compile-verified
//
#include <hip/hip_runtime.h>

// ---------------------------------------------------------------------------
// Affinity kernel for MI455X (gfx1250, wave32)
//   M[a,b] = sum_h w2[h]*relu( (X@Wx^T)[a,h] + (Y@Wy^T)[b,h] + b1[h] ) + b2
// where Wx = W1[:, :C]@Wsr, Wy = W1[:, C:]@Wtg  (weight folding).
// Stages 1/2 are exact-fp32 WMMA GEMMs (V_WMMA_F32_16X16X4_F32).
// Stage 3 (dominant, ~0.8 GFLOP) is VALU-bound; LDS-tiled with 4x4 register
// blocking, bank-conflict-free transposed tiles, float4 LDS loads.
// ---------------------------------------------------------------------------

typedef __attribute__((ext_vector_type(2))) float v2f;
typedef __attribute__((ext_vector_type(8))) float v8f;

#define N1_ 1024
#define N2_ 1024
#define C_  128
#define H_  256

// ---------------------------------------------------------------------------
// Stage 1: fold weights.  Wf[mat] = W1[:, mat*C : (mat+1)*C] @ (mat? Wtg : Wsr)
// Output Wf: [2][256][128] row-major.  One wave per 16x16 output tile.
// ---------------------------------------------------------------------------
__global__ void fold_weights_wmma(const float* __restrict__ W1,
                                  const float* __restrict__ Wsr,
                                  const float* __restrict__ Wtg,
                                  float* __restrict__ Wf) {
  const int lane = threadIdx.x;     // 0..31, wave32
  const int half = lane >> 4;       // 0 or 1 (K half of the frag)
  const int lr   = lane & 15;
  const int nb   = blockIdx.x * 16; // N base over C_=128
  const int mb   = blockIdx.y * 16; // M base over H_=256
  const int mat  = blockIdx.z;      // 0 -> Wx, 1 -> Wy

  const float* A = W1 + mat * C_;             // A[j][c] = W1[j][mat*C + c]
  const float* B = mat ? Wtg : Wsr;           // [128][128] row-major
  float*       O = Wf + mat * (H_ * C_);

  v8f acc = {};
  for (int kk = 0; kk < C_; kk += 4) {
    const int k0 = kk + 2 * half;
    v2f a, b;
    a.x = A[(mb + lr) * H_ + k0];
    a.y = A[(mb + lr) * H_ + k0 + 1];
    b.x = B[(k0    ) * C_ + nb + lr];
    b.y = B[(k0 + 1) * C_ + nb + lr];
    acc = __builtin_amdgcn_wmma_f32_16x16x4_f32(false, a, false, b,
                                                (short)0, acc, false, false);
  }
#pragma unroll
  for (int r = 0; r < 8; ++r) {
    const int row = mb + half * 8 + r;        // D: lane<16 rows 0..7, else 8..15
    O[row * C_ + nb + lr] = acc[r];
  }
}

// ---------------------------------------------------------------------------
// Stage 2: projections.
//   Xh[a,j] = sum_k X[a,k]*Wx[j,k]          -> XhYh[0]
//   Yh[b,j] = sum_k Y[b,k]*Wy[j,k] + b1[j]  -> XhYh[1]   (b1 folded in)
// XhYh: [2][1024][256] row-major.  One wave per 16x16 tile, K=128 in steps of 4.
// ---------------------------------------------------------------------------
__global__ void proj_wmma(const float* __restrict__ X,
                          const float* __restrict__ Y,
                          const float* __restrict__ Wf,
                          const float* __restrict__ b1,
                          float* __restrict__ XhYh) {
  const int lane = threadIdx.x;
  const int half = lane >> 4;
  const int lr   = lane & 15;
  const int nb   = blockIdx.x * 16;   // over H_=256
  const int mb   = blockIdx.y * 16;   // over rows (1024)
  const int mat  = blockIdx.z;        // 0 -> X path, 1 -> Y path

  const float* A = mat ? Y : X;               // [1024][128] row-major
  const float* W = Wf + mat * (H_ * C_);      // B[k][n] = W[n][k]
  float*       O = XhYh + mat * (N1_ * H_);

  v8f acc = {};
  for (int kk = 0; kk < C_; kk += 4) {
    const int k0 = kk + 2 * half;
    v2f a, b;
    a.x = A[(mb + lr) * C_ + k0];
    a.y = A[(mb + lr) * C_ + k0 + 1];
    b.x = W[(nb + lr) * C_ + k0];             // consecutive -> b64 load
    b.y = W[(nb + lr) * C_ + k0 + 1];
    acc = __builtin_amdgcn_wmma_f32_16x16x4_f32(false, a, false, b,
                                                (short)0, acc, false, false);
  }
#pragma unroll
  for (int r = 0; r < 8; ++r) {
    const int row  = mb + half * 8 + r;
    const int col  = nb + lr;
    const float bv = (mat == 1) ? b1[col] : 0.0f;
    O[row * H_ + col] = acc[r] + bv;
  }
}

// ---------------------------------------------------------------------------
// Stage 3: epilogue (VALU-bound).  64x64 output tile per 256-thread block,
// 4x4 register blocking per thread, h processed in 2 chunks of 128 with
// transposed LDS tiles Xs[h][a], Ys[h][b] (bank = a/b index -> conflict-free,
// 16B-aligned float4 ds loads).
// ---------------------------------------------------------------------------
#define TA 64
#define TB 64
#define HCHUNK 128

__global__ __launch_bounds__(256) void affinity_epilogue(
    const float* __restrict__ XhYh,
    const float* __restrict__ W2,
    const float* __restrict__ b2,
    float* __restrict__ M) {
  __shared__ float Xs[HCHUNK * TA];   // 32 KB, [h][a]
  __shared__ float Ys[HCHUNK * TB];   // 32 KB, [h][b]

  const float* Xh = XhYh;
  const float* Yh = XhYh + N1_ * H_;

  const int tid = threadIdx.x;
  const int tx  = tid & 15;           // b-group
  const int ty  = tid >> 4;           // a-group
  const int a0  = blockIdx.y * TA;
  const int b0  = blockIdx.x * TB;

  float acc[16];
#pragma unroll
  for (int i = 0; i < 16; ++i) acc[i] = 0.0f;

  for (int hb = 0; hb < H_; hb += HCHUNK) {
    // Stage tiles transposed: lane's LDS bank = row index -> no conflicts.
    for (int idx = tid; idx < TA * (HCHUNK / 4); idx += 256) {
      const int a  = idx & (TA - 1);
      const int h4 = idx >> 6;        // TA == 64
      const float4 v =
          *reinterpret_cast<const float4*>(&Xh[(a0 + a) * H_ + hb + 4 * h4]);
      Xs[(4 * h4 + 0) * TA + a] = v.x;
      Xs[(4 * h4 + 1) * TA + a] = v.y;
      Xs[(4 * h4 + 2) * TA + a] = v.z;
      Xs[(4 * h4 + 3) * TA + a] = v.w;
    }
    for (int idx = tid; idx < TB * (HCHUNK / 4); idx += 256) {
      const int b  = idx & (TB - 1);
      const int h4 = idx >> 6;
      const float4 v =
          *reinterpret_cast<const float4*>(&Yh[(b0 + b) * H_ + hb + 4 * h4]);
      Ys[(4 * h4 + 0) * TB + b] = v.x;
      Ys[(4 * h4 + 1) * TB + b] = v.y;
      Ys[(4 * h4 + 2) * TB + b] = v.z;
      Ys[(4 * h4 + 3) * TB + b] = v.w;
    }
    __syncthreads();

#pragma unroll 8
    for (int h = 0; h < HCHUNK; ++h) {
      const float w = W2[hb + h];     // uniform -> scalar load
      const float4 xa =
          *reinterpret_cast<const float4*>(&Xs[h * TA + ty * 4]);
      const float4 yb =
          *reinterpret_cast<const float4*>(&Ys[h * TB + tx * 4]);
      const float xv[4] = {xa.x, xa.y, xa.z, xa.w};
      const float yv[4] = {yb.x, yb.y, yb.z, yb.w};
#pragma unroll
      for (int i = 0; i < 4; ++i)
#pragma unroll
        for (int j = 0; j < 4; ++j)
          acc[i * 4 + j] =
              fmaf(w, fmaxf(xv[i] + yv[j], 0.0f), acc[i * 4 + j]);
    }
    __syncthreads();
  }

  const float bias2 = b2[0];
#pragma unroll
  for (int i = 0; i < 4; ++i) {
    float4 o;
    o.x = acc[i * 4 + 0] + bias2;
    o.y = acc[i * 4 + 1] + bias2;
    o.z = acc[i * 4 + 2] + bias2;
    o.w = acc[i * 4 + 3] + bias2;
    *reinterpret_cast<float4*>(&M[(a0 + ty * 4 + i) * N2_ + b0 + tx * 4]) = o;
  }
}

// ---------------------------------------------------------------------------
// Host launcher
// ---------------------------------------------------------------------------
extern "C" void kernel_launch(void* const* d_in, const int* in_sizes, int n_in,
                              void* d_out, int out_size, void* d_ws,
                              size_t ws_size, hipStream_t stream) {
  (void)in_sizes; (void)n_in; (void)out_size; (void)ws_size;

  const float* X   = (const float*)d_in[0];  // [1024,128]
  const float* Y   = (const float*)d_in[1];  // [1024,128]
  const float* Wsr = (const float*)d_in[2];  // [128,128]
  const float* Wtg = (const float*)d_in[3];  // [128,128]
  const float* W1  = (const float*)d_in[4];  // [256,256]
  const float* b1  = (const float*)d_in[5];  // [256]
  const float* W2  = (const float*)d_in[6];  // [1,256]
  const float* b2  = (const float*)d_in[7];  // [1]
  float* M = (float*)d_out;                  // [1024,1024]

  float* ws   = (float*)d_ws;
  float* Wf   = ws;                  // [2][256][128]  = 32768*2 floats
  float* XhYh = ws + 2 * H_ * C_;    // [2][1024][256] = 262144*2 floats
  // total workspace: 589824 floats = 2.25 MB

  fold_weights_wmma<<<dim3(C_ / 16, H_ / 16, 2), 32, 0, stream>>>(W1, Wsr, Wtg,
                                                                  Wf);
  proj_wmma<<<dim3(H_ / 16, N1_ / 16, 2), 32, 0, stream>>>(X, Y, Wf, b1, XhYh);
  affinity_epilogue<<<dim3(N2_ / TB, N1_ / TA), 256, 0, stream>>>(XhYh, W2, b2,
                                                                  M);
}